// SparseMinCostFlow_20444044329127
// MI455X (gfx1250) — compile-verified
//
#include <hip/hip_runtime.h>
#include <stdint.h>

// Sparse min-cost-flow relaxation for MI455X (gfx1250).
// Memory-bound scatter/gather workload: no matrix structure -> no WMMA.
// CDNA5 paths used instead:
//   * global_load_async_to_lds_b128 + s_wait_asynccnt (double-buffered stream staging)
//   * global_atomic_add_f32 (relaxed, agent scope) into L2-resident 2MB inflow array
//   * non-temporal (TH=NT) loads on 128MB/iter edge streams to protect flow in 192MB L2
//   * wave32 ballot for the convergence flag

typedef float f32x4 __attribute__((ext_vector_type(4)));
typedef int   i32x4 __attribute__((ext_vector_type(4)));

#define FLOW_THRESHOLD 1.0e-4f
#define FLOW_ITERS     10

#define SC_BLOCK 256
#define SC_WAVES (SC_BLOCK / 32)
#define TILE     128   // edges per wave-tile: one b128 async op per lane per array

__device__ __forceinline__ bool wave_any(bool p) {
  return __builtin_amdgcn_ballot_w32(p) != 0u;
}

// Raw 32-bit LDS byte address for async-copy destination operand.
__device__ __forceinline__ unsigned lds_u32(const void* p) {
  return (unsigned)(uintptr_t)(__attribute__((address_space(3))) const void*)p;
}

// GLOBAL_LOAD_ASYNC_TO_LDS_B128, GVS mode: saddr = 64-bit base, vaddr = 32-bit byte offset.
// Each lane moves 16B from global memory to the LDS address in lds_addr; tracked by ASYNCcnt.
__device__ __forceinline__ void async_b128(unsigned lds_addr, const void* base, unsigned byte_off) {
  asm volatile("global_load_async_to_lds_b128 %0, %1, %2"
               :: "v"(lds_addr), "v"(byte_off), "s"(base)
               : "memory");
}

// Same, but non-temporal: pure stream, do not allocate in L2 (protects the
// L2-resident flow array from eviction by the 128MB/iter edge_dst stream).
__device__ __forceinline__ void async_b128_nt(unsigned lds_addr, const void* base, unsigned byte_off) {
  asm volatile("global_load_async_to_lds_b128 %0, %1, %2 th:TH_LOAD_NT"
               :: "v"(lds_addr), "v"(byte_off), "s"(base)
               : "memory");
}

// ---------------------------------------------------------------------------
// Scatter-add: inflow[edge_dst[e]] += flow[e].
// Streams (flow, edge_dst) are staged into LDS with double-buffered async b128
// copies; each lane consumes exactly the 16B it staged, so a wave-level
// s_wait_asynccnt suffices (no barrier). Atomics land in the L2-resident
// 2MB inflow array. flow is loaded default-RT (expected L2 hit from the
// previous iteration's update store); edge_dst is loaded TH_LOAD_NT.
// ---------------------------------------------------------------------------
__global__ void __launch_bounds__(SC_BLOCK)
k_scatter(const float* __restrict__ flow, const int* __restrict__ edge_dst,
          float* __restrict__ inflow, const int* __restrict__ flags,
          int it, int n_edges) {
  if (flags[it] == 0) return;

  __shared__ float s_flow[2][SC_WAVES][TILE];
  __shared__ int   s_dst [2][SC_WAVES][TILE];

  const int lane = threadIdx.x & 31;
  const int wid  = threadIdx.x >> 5;
  const int gw   = blockIdx.x * SC_WAVES + wid;   // global wave id
  const int nw   = gridDim.x * SC_WAVES;          // total waves
  const int ntiles = n_edges / TILE;

  const unsigned lflow[2] = { lds_u32(&s_flow[0][wid][lane * 4]),
                              lds_u32(&s_flow[1][wid][lane * 4]) };
  const unsigned ldst[2]  = { lds_u32(&s_dst[0][wid][lane * 4]),
                              lds_u32(&s_dst[1][wid][lane * 4]) };
  const unsigned laneB = (unsigned)lane * 16u;

  int t = gw, buf = 0;
  if (t < ntiles) {
    unsigned off = (unsigned)t * (TILE * 4u) + laneB;
    async_b128(lflow[0], flow, off);
    async_b128_nt(ldst[0], edge_dst, off);
  }
  for (; t < ntiles; t += nw) {
    const int tn = t + nw;
    if (tn < ntiles) {
      unsigned off = (unsigned)tn * (TILE * 4u) + laneB;
      async_b128(lflow[buf ^ 1], flow, off);
      async_b128_nt(ldst[buf ^ 1], edge_dst, off);
      asm volatile("s_wait_asynccnt 2" ::: "memory");  // current tile done, next in flight
    } else {
      asm volatile("s_wait_asynccnt 0" ::: "memory");
    }
#pragma unroll
    for (int k = 0; k < 4; ++k) {
      const int   d = s_dst[buf][wid][lane * 4 + k];
      const float f = s_flow[buf][wid][lane * 4 + k];
      __hip_atomic_fetch_add(&inflow[d], f, __ATOMIC_RELAXED, __HIP_MEMORY_SCOPE_AGENT);
    }
    buf ^= 1;
  }
  // tail (n_edges not multiple of TILE)
  for (int e = ntiles * TILE + gw * 32 + lane; e < n_edges; e += nw * 32) {
    __hip_atomic_fetch_add(&inflow[edge_dst[e]], flow[e],
                           __ATOMIC_RELAXED, __HIP_MEMORY_SCOPE_AGENT);
  }
}

// ---------------------------------------------------------------------------
// flags[0..FLOW_ITERS]: flags[it] != 0 means iteration `it` must run.
// ---------------------------------------------------------------------------

__global__ void k_zero_flags(int* flags) {
  if (threadIdx.x < 16) flags[threadIdx.x] = 0;
}

// Copy values -> flow (in d_out) and compute flags[0] = any(values > threshold)
// (reference: first cond checks flow(=values) - prev(=0) > threshold).
__global__ void k_init(const float* __restrict__ values, float* __restrict__ flow,
                       int* __restrict__ flags, int n4, int n_edges) {
  int i = blockIdx.x * blockDim.x + threadIdx.x;
  const int stride = gridDim.x * blockDim.x;
  bool hit = false;
  for (; i < n4; i += stride) {
    f32x4 v = __builtin_nontemporal_load((const f32x4*)values + i);
    ((f32x4*)flow)[i] = v;  // temporal: scatter re-reads this from L2 next
    hit |= (v.x > FLOW_THRESHOLD) || (v.y > FLOW_THRESHOLD) ||
           (v.z > FLOW_THRESHOLD) || (v.w > FLOW_THRESHOLD);
  }
  for (int e = n4 * 4 + blockIdx.x * blockDim.x + threadIdx.x; e < n_edges; e += stride) {
    float v = values[e];
    flow[e] = v;
    hit |= (v > FLOW_THRESHOLD);
  }
  if (wave_any(hit) && (threadIdx.x & 31) == 0) flags[0] = 1;
}

__global__ void k_zero_inflow(float* __restrict__ inflow, const int* __restrict__ flags,
                              int it, int n_nodes) {
  if (flags[it] == 0) return;
  int i = blockIdx.x * blockDim.x + threadIdx.x;
  const int stride = gridDim.x * blockDim.x;
  for (; i < n_nodes; i += stride) inflow[i] = 0.0f;
}

// adjusted[n] = max(inflow[n] - demand[n], 0), in place over inflow.
__global__ void k_adjust(float* __restrict__ inflow, const float* __restrict__ demands,
                         const int* __restrict__ flags, int it, int n_nodes) {
  if (flags[it] == 0) return;
  int i = blockIdx.x * blockDim.x + threadIdx.x;
  const int stride = gridDim.x * blockDim.x;
  for (; i < n_nodes; i += stride)
    inflow[i] = fmaxf(inflow[i] - demands[i], 0.0f);
}

// flow[e] = values[e] * adjusted[edge_src[e]], in place; set flags[it+1] if
// any lane sees new - old > threshold. values/edge_src are NT streams; the
// adjusted gather is a plain load (2MB, L2-resident); old flow load and new
// flow store are default-RT so flow stays L2-resident across kernels.
__global__ void k_update(const float* __restrict__ values, const int* __restrict__ edge_src,
                         const float* __restrict__ adjusted, float* __restrict__ flow,
                         int* __restrict__ flags, int it, int n4, int n_edges) {
  if (flags[it] == 0) return;
  int i = blockIdx.x * blockDim.x + threadIdx.x;
  const int stride = gridDim.x * blockDim.x;
  bool hit = false;
  for (; i < n4; i += stride) {
    f32x4 v = __builtin_nontemporal_load((const f32x4*)values + i);
    i32x4 s = __builtin_nontemporal_load((const i32x4*)edge_src + i);
    f32x4 old = ((const f32x4*)flow)[i];
    f32x4 nf;
    nf.x = v.x * adjusted[s.x];
    nf.y = v.y * adjusted[s.y];
    nf.z = v.z * adjusted[s.z];
    nf.w = v.w * adjusted[s.w];
    hit |= ((nf.x - old.x) > FLOW_THRESHOLD) || ((nf.y - old.y) > FLOW_THRESHOLD) ||
           ((nf.z - old.z) > FLOW_THRESHOLD) || ((nf.w - old.w) > FLOW_THRESHOLD);
    ((f32x4*)flow)[i] = nf;
  }
  for (int e = n4 * 4 + blockIdx.x * blockDim.x + threadIdx.x; e < n_edges; e += stride) {
    float old = flow[e];
    float nf  = values[e] * adjusted[edge_src[e]];
    hit |= ((nf - old) > FLOW_THRESHOLD);
    flow[e] = nf;
  }
  if (wave_any(hit) && (threadIdx.x & 31) == 0) flags[it + 1] = 1;
}

// ---------------------------------------------------------------------------

extern "C" void kernel_launch(void* const* d_in, const int* in_sizes, int n_in,
                              void* d_out, int out_size, void* d_ws, size_t ws_size,
                              hipStream_t stream) {
  const float* values   = (const float*)d_in[0];
  const float* demands  = (const float*)d_in[1];
  const int*   edge_src = (const int*)d_in[2];
  const int*   edge_dst = (const int*)d_in[3];
  float*       flow     = (float*)d_out;   // flow evolves in place here

  const int n_edges = in_sizes[0];
  const int n_nodes = in_sizes[1];
  const int n4      = n_edges >> 2;

  // workspace: [inflow/adjusted: n_nodes floats][flags: 16 ints]
  float* inflow = (float*)d_ws;
  int*   flags  = (int*)((char*)d_ws + (((size_t)n_nodes * 4 + 255) & ~(size_t)255));

  const dim3 B(256);
  const int gInit = 8192;
  const int gNode = (n_nodes + 255) / 256;
  const int gScat = 2048;
  const int gUpd  = 8192;

  k_zero_flags<<<1, 32, 0, stream>>>(flags);
  k_init<<<gInit, B, 0, stream>>>(values, flow, flags, n4, n_edges);

  for (int it = 0; it < FLOW_ITERS; ++it) {
    k_zero_inflow<<<gNode, B, 0, stream>>>(inflow, flags, it, n_nodes);
    k_scatter<<<gScat, B, 0, stream>>>(flow, edge_dst, inflow, flags, it, n_edges);
    k_adjust<<<gNode, B, 0, stream>>>(inflow, demands, flags, it, n_nodes);
    k_update<<<gUpd, B, 0, stream>>>(values, edge_src, inflow, flow, flags, it, n4, n_edges);
  }
}